// RBFKernelFunction_47614007443547
// MI455X (gfx1250) — compile-verified
//
#include <hip/hip_runtime.h>

typedef __attribute__((ext_vector_type(2))) float v2f;
typedef __attribute__((ext_vector_type(8))) float v8f;
typedef __attribute__((ext_vector_type(4))) float f4;

#define N_PTS 512
#define M_PTS 512
#define DD    16
#define WROW  (M_PTS * (DD + 1))   // 8704 floats per output row

__global__ __launch_bounds__(256)
void rbf_full_k(const float* __restrict__ X,
                const float* __restrict__ Y,
                const float* __restrict__ theta,
                float* __restrict__ out)
{
    __shared__ float Xl[256];        // 16x16 X tile, row-major
    __shared__ float Yl[256];        // 16x16 Y tile, row-major
    __shared__ float it2[16];        // 1/theta^2
    __shared__ float sxl[16];        // sum_d X[i,d]^2 * it2[d]
    __shared__ float syl[16];
    __shared__ float Kl[256];        // K tile (il*16 + jl)
    __shared__ float St[16 * 320];   // scaled diffs: St[il*320 + jl*20 + d], pad=20

    const unsigned t  = threadIdx.x;
    const unsigned i0 = blockIdx.y * 16u;
    const unsigned j0 = blockIdx.x * 16u;

    // ---- stage tiles (coalesced 1KB each) ----
    Xl[t] = X[i0 * DD + t];
    Yl[t] = Y[j0 * DD + t];
    if (t < 16) { float th = theta[t]; it2[t] = 1.0f / (th * th); }
    __syncthreads();

    // ---- per-pair scaled diffs -> LDS (bank-conflict-free, stride 20) ----
    {
        const unsigned il = t >> 4, jl = t & 15;
        float s[DD];
        #pragma unroll
        for (int d = 0; d < DD; ++d) {
            float df = Xl[il * 16 + d] - Yl[jl * 16 + d];
            s[d] = df * it2[d];
        }
        float* sp = &St[il * 320 + jl * 20];
        #pragma unroll
        for (int d = 0; d < DD; d += 4) {
            f4 v; v.x = s[d]; v.y = s[d + 1]; v.z = s[d + 2]; v.w = s[d + 3];
            *(f4*)(sp + d) = v;
        }
    }
    // ---- row norms for r2 decomposition ----
    if (t < 16) {
        float acc = 0.f;
        #pragma unroll
        for (int d = 0; d < DD; ++d) { float v = Xl[t * 16 + d]; acc += v * v * it2[d]; }
        sxl[t] = acc;
    } else if (t < 32) {
        const unsigned j = t - 16;
        float acc = 0.f;
        #pragma unroll
        for (int d = 0; d < DD; ++d) { float v = Yl[j * 16 + d]; acc += v * v * it2[d]; }
        syl[j] = acc;
    }
    __syncthreads();

    // ---- wave 0: cross term via WMMA f32 16x16x4 (K=16 -> 4 steps), then K ----
    if (t < 32) {
        const unsigned m = t & 15, h = t >> 4;
        v8f c = {};
        #pragma unroll
        for (int kk = 0; kk < 4; ++kk) {
            // A (16x4, M=i rows, scaled by it2): lanes 0-15 hold K=kk*4+{0,1},
            // lanes 16-31 hold K=kk*4+{2,3} in the two VGPRs.
            v2f a, b;
            const int ka = kk * 4 + 2 * (int)h;
            a.x = Xl[m * 16 + ka]     * it2[ka];
            a.y = Xl[m * 16 + ka + 1] * it2[ka + 1];
            // B (4x16, K rows striped across lane halves per VGPR)
            const int kb = kk * 4 + (int)h;
            b.x = Yl[m * 16 + kb];
            b.y = Yl[m * 16 + kb + 2];
            c = __builtin_amdgcn_wmma_f32_16x16x4_f32(
                    false, a, false, b, (short)0, c, false, false);
        }
        // C/D layout: VGPR r, lane l -> i = r + 8*(l/16), j = l%16
        #pragma unroll
        for (int r = 0; r < 8; ++r) {
            const unsigned il = (unsigned)r + 8u * h, jl = m;
            float r2 = sxl[il] + syl[jl] - 2.0f * c[r];
            float k  = __expf(-0.5f * r2);
            Kl[il * 16 + jl] = k;
            __builtin_nontemporal_store(
                k, &out[(size_t)(i0 + il) * WROW + (j0 + jl)]);
        }
    }
    __syncthreads();

    // ---- KFG: rows i, cols [M + j0*16, +256): 1KB contiguous per row ----
    {
        const unsigned jl = t >> 4, d = t & 15;
        #pragma unroll 4
        for (unsigned il = 0; il < 16; ++il) {
            float k = Kl[il * 16 + jl];
            float s = St[il * 320 + jl * 20 + d];
            __builtin_nontemporal_store(
                k * s, &out[(size_t)(i0 + il) * WROW + M_PTS + j0 * DD + t]);
        }
    }
    // ---- KGF: rows N + i*16 + a, cols [j0, j0+16): 16x 64B runs per iter ----
    {
        const unsigned a = t >> 4, jl = t & 15;
        #pragma unroll 4
        for (unsigned il = 0; il < 16; ++il) {
            float k  = Kl[il * 16 + jl];
            float sa = St[il * 320 + jl * 20 + a];
            size_t row = (size_t)N_PTS + (size_t)(i0 + il) * DD + a;
            __builtin_nontemporal_store(-k * sa, &out[row * WROW + j0 + jl]);
        }
    }
    // ---- KGG: 65536 floats/tile, float4 stores, 4KB per block-wide store ----
    #pragma unroll 2
    for (unsigned it = 0; it < 64; ++it) {
        const unsigned g  = it * 1024u + t * 4u;
        const unsigned il = g >> 12;
        const unsigned a  = (g >> 8) & 15u;
        const unsigned jl = (g >> 4) & 15u;
        const unsigned b0 = g & 15u;              // multiple of 4
        const float k  = Kl[il * 16 + jl];
        const float* sp = &St[il * 320 + jl * 20];
        const float sa = sp[a];
        const float ia = it2[a];
        f4 v;
        v.x = k * (((a == b0 + 0u) ? ia : 0.f) - sa * sp[b0 + 0]);
        v.y = k * (((a == b0 + 1u) ? ia : 0.f) - sa * sp[b0 + 1]);
        v.z = k * (((a == b0 + 2u) ? ia : 0.f) - sa * sp[b0 + 2]);
        v.w = k * (((a == b0 + 3u) ? ia : 0.f) - sa * sp[b0 + 3]);
        size_t row = (size_t)N_PTS + (size_t)(i0 + il) * DD + a;
        float* dst = &out[row * WROW + M_PTS + (size_t)(j0 + jl) * DD + b0];
        __builtin_nontemporal_store(v, (f4*)dst);
    }
}

extern "C" void kernel_launch(void* const* d_in, const int* in_sizes, int n_in,
                              void* d_out, int out_size, void* d_ws, size_t ws_size,
                              hipStream_t stream) {
    (void)in_sizes; (void)n_in; (void)out_size; (void)d_ws; (void)ws_size;
    const float* X     = (const float*)d_in[0];
    const float* Y     = (const float*)d_in[1];
    const float* theta = (const float*)d_in[2];
    float* out = (float*)d_out;
    dim3 grid(M_PTS / 16, N_PTS / 16);  // 32 x 32 = 1024 tiles
    rbf_full_k<<<grid, 256, 0, stream>>>(X, Y, theta, out);
}